// GCN_2903397892892
// MI455X (gfx1250) — compile-verified
//
#include <hip/hip_runtime.h>
#include <math.h>

#define NN 100000      // nodes
#define NE 1600000     // edges
#define FIN 128
#define HID 64
#define NCLS 10
#define ROW_TILES 6250 // NN / 16 exactly

typedef __attribute__((ext_vector_type(2))) float v2f;
typedef __attribute__((ext_vector_type(8))) float v8f;

// ---------------- utility kernels ----------------
__global__ void k_fill(float* __restrict__ p, int n, float v) {
    int i = blockIdx.x * blockDim.x + threadIdx.x;
    if (i < n) p[i] = v;
}

__global__ void k_deg(const int* __restrict__ dst, float* __restrict__ deg) {
    int e = blockIdx.x * blockDim.x + threadIdx.x;
    if (e < NE) atomicAdd(&deg[dst[e]], 1.0f);
}

__global__ void k_dis(float* __restrict__ deg) {
    int i = blockIdx.x * blockDim.x + threadIdx.x;
    if (i < NN) deg[i] = rsqrtf(deg[i]);  // deg >= 1 always (self-loop)
}

__global__ void k_norm(const int* __restrict__ src, const int* __restrict__ dst,
                       const float* __restrict__ dis, float* __restrict__ norm) {
    int e = blockIdx.x * blockDim.x + threadIdx.x;
    if (e < NE) norm[e] = dis[src[e]] * dis[dst[e]];
}

// ---------------- GEMM1: H1[NN,64] = X[NN,128] @ W1[128,64] (fp32 WMMA) -------------
// 128 threads = 4 waves; each wave computes one 16-row tile across all 64 cols.
__global__ void __launch_bounds__(128)
k_gemm1(const float* __restrict__ x, const float* __restrict__ W1,
        float* __restrict__ h1) {
    // W1 transposed in LDS, row pitch 132 (132 % 64 == 4 -> conflict-free frag loads)
    __shared__ float wT[HID * 132];
    int t = threadIdx.x;
    for (int i = t; i < FIN * HID; i += 128) {
        int k = i >> 6, c = i & 63;          // W1[k][c]
        wT[c * 132 + k] = W1[i];
    }
    __syncthreads();

    int tile = blockIdx.x * 4 + (t >> 5);
    if (tile >= ROW_TILES) return;           // wave-uniform: EXEC all-1s for WMMA
    int lane = t & 31;
    int m = lane & 15, half = lane >> 4;

    const float* arow = x + (size_t)(tile * 16 + m) * FIN + 2 * half;
    v8f a0 = {}, a1 = {}, a2 = {}, a3 = {};

    #pragma unroll 4
    for (int k0 = 0; k0 < FIN; k0 += 4) {
        v2f a = *(const v2f*)(arow + k0);                       // A[m][k0+2h+{0,1}]
        const float* bp = wT + m * 132 + k0 + 2 * half;         // B[k][n=m], col tile stride 16*132
        v2f b0 = *(const v2f*)(bp);
        v2f b1 = *(const v2f*)(bp + 16 * 132);
        v2f b2 = *(const v2f*)(bp + 32 * 132);
        v2f b3 = *(const v2f*)(bp + 48 * 132);
        a0 = __builtin_amdgcn_wmma_f32_16x16x4_f32(false, a, false, b0, (short)0, a0, false, false);
        a1 = __builtin_amdgcn_wmma_f32_16x16x4_f32(false, a, false, b1, (short)0, a1, false, false);
        a2 = __builtin_amdgcn_wmma_f32_16x16x4_f32(false, a, false, b2, (short)0, a2, false, false);
        a3 = __builtin_amdgcn_wmma_f32_16x16x4_f32(false, a, false, b3, (short)0, a3, false, false);
    }

    float* out = h1 + (size_t)(tile * 16 + 8 * half) * HID + m; // row = v + 8*half, col = c*16 + m
    #pragma unroll
    for (int v = 0; v < 8; ++v) {
        out[v * HID +  0] = a0[v];
        out[v * HID + 16] = a1[v];
        out[v * HID + 32] = a2[v];
        out[v * HID + 48] = a3[v];
    }
}

// ---------------- GEMM2: H2[NN,10] = Z1[NN,64] @ W2[64,10] (padded to 16 cols) ------
__global__ void __launch_bounds__(128)
k_gemm2(const float* __restrict__ z1, const float* __restrict__ W2,
        float* __restrict__ h2) {
    __shared__ float wT[16 * 68];            // pitch 68 (68 % 64 == 4)
    int t = threadIdx.x;
    for (int i = t; i < 16 * HID; i += 128) {
        int k = i >> 4, c = i & 15;
        wT[c * 68 + k] = (c < NCLS) ? W2[k * NCLS + c] : 0.0f;
    }
    __syncthreads();

    int tile = blockIdx.x * 4 + (t >> 5);
    if (tile >= ROW_TILES) return;
    int lane = t & 31;
    int m = lane & 15, half = lane >> 4;

    const float* arow = z1 + (size_t)(tile * 16 + m) * HID + 2 * half;
    v8f acc = {};
    #pragma unroll
    for (int k0 = 0; k0 < HID; k0 += 4) {
        v2f a = *(const v2f*)(arow + k0);
        v2f b = *(const v2f*)(wT + m * 68 + k0 + 2 * half);
        acc = __builtin_amdgcn_wmma_f32_16x16x4_f32(false, a, false, b, (short)0, acc, false, false);
    }
    if (m < NCLS) {
        float* out = h2 + (size_t)(tile * 16 + 8 * half) * NCLS + m;
        #pragma unroll
        for (int v = 0; v < 8; ++v) out[v * NCLS] = acc[v];
    }
}

// ---------------- edge scatter-add (layer 1, F=64: 64 lanes per edge) ---------------
__global__ void k_scatter1(const int* __restrict__ src, const int* __restrict__ dst,
                           const float* __restrict__ norm, const float* __restrict__ h1,
                           float* __restrict__ agg) {
    int g = blockIdx.x * blockDim.x + threadIdx.x;   // NE*64 = 102.4M < 2^31
    int e = g >> 6, f = g & 63;
    if (e >= NE) return;
    int s = src[e], d = dst[e];
    atomicAdd(&agg[(size_t)d * HID + f], h1[(size_t)s * HID + f] * norm[e]);
}

// ---------------- edge scatter-add (layer 2, F=10: 16 lanes per edge) ---------------
__global__ void k_scatter2(const int* __restrict__ src, const int* __restrict__ dst,
                           const float* __restrict__ norm, const float* __restrict__ h2,
                           float* __restrict__ out) {
    int g = blockIdx.x * blockDim.x + threadIdx.x;   // NE*16 = 25.6M
    int e = g >> 4, f = g & 15;
    if (e >= NE || f >= NCLS) return;
    int s = src[e], d = dst[e];
    atomicAdd(&out[(size_t)d * NCLS + f], h2[(size_t)s * NCLS + f] * norm[e]);
}

// ---------------- self-loop + bias + relu (layer 1), in-place over agg --------------
__global__ void k_post1(float* __restrict__ agg, const float* __restrict__ h1,
                        const float* __restrict__ dis, const float* __restrict__ b1) {
    int i = blockIdx.x * blockDim.x + threadIdx.x;
    if (i >= NN * HID) return;
    int node = i >> 6, f = i & 63;
    float di = dis[node];
    float v = agg[i] + h1[i] * di * di + b1[f];
    agg[i] = v > 0.0f ? v : 0.0f;
}

// ---------------- self-loop + bias + log_softmax (layer 2), in-place on d_out -------
__global__ void k_post2(float* __restrict__ out, const float* __restrict__ h2,
                        const float* __restrict__ dis, const float* __restrict__ b2) {
    int i = blockIdx.x * blockDim.x + threadIdx.x;
    if (i >= NN) return;
    float d2 = dis[i] * dis[i];
    float tv[NCLS];
    float mx = -3.402823466e38f;
    #pragma unroll
    for (int f = 0; f < NCLS; ++f) {
        float v = out[(size_t)i * NCLS + f] + h2[(size_t)i * NCLS + f] * d2 + b2[f];
        tv[f] = v;
        mx = fmaxf(mx, v);
    }
    float s = 0.0f;
    #pragma unroll
    for (int f = 0; f < NCLS; ++f) s += expf(tv[f] - mx);
    float lse = mx + logf(s);
    #pragma unroll
    for (int f = 0; f < NCLS; ++f) out[(size_t)i * NCLS + f] = tv[f] - lse;
}

extern "C" void kernel_launch(void* const* d_in, const int* in_sizes, int n_in,
                              void* d_out, int out_size, void* d_ws, size_t ws_size,
                              hipStream_t stream) {
    const float* x  = (const float*)d_in[0];
    const int*   ei = (const int*)d_in[1];     // edge_index [2, NE]
    const float* W1 = (const float*)d_in[2];
    const float* b1 = (const float*)d_in[3];
    const float* W2 = (const float*)d_in[4];
    const float* b2 = (const float*)d_in[5];
    const int* src = ei;
    const int* dst = ei + NE;
    float* out = (float*)d_out;

    // workspace layout (floats): dis[NN] | h1[NN*64] | agg1[NN*64] | h2[NN*10] | norm[NE]
    float* ws   = (float*)d_ws;
    float* dis  = ws;
    float* h1   = dis + NN;
    float* agg1 = h1 + (size_t)NN * HID;
    float* h2   = agg1 + (size_t)NN * HID;
    float* norm = h2 + (size_t)NN * NCLS;

    const int B = 256;
    // init: deg=1 (self loop), agg1=0, out=0
    k_fill<<<(NN + B - 1) / B, B, 0, stream>>>(dis, NN, 1.0f);
    k_fill<<<(NN * HID + B - 1) / B, B, 0, stream>>>(agg1, NN * HID, 0.0f);
    k_fill<<<(NN * NCLS + B - 1) / B, B, 0, stream>>>(out, NN * NCLS, 0.0f);

    // degrees -> deg_inv_sqrt -> per-edge norm
    k_deg<<<(NE + B - 1) / B, B, 0, stream>>>(dst, dis);
    k_dis<<<(NN + B - 1) / B, B, 0, stream>>>(dis);
    k_norm<<<(NE + B - 1) / B, B, 0, stream>>>(src, dst, dis, norm);

    // layer 1
    k_gemm1<<<(ROW_TILES + 3) / 4, 128, 0, stream>>>(x, W1, h1);
    k_scatter1<<<(NE * 64) / B, B, 0, stream>>>(src, dst, norm, h1, agg1);
    k_post1<<<(NN * HID + B - 1) / B, B, 0, stream>>>(agg1, h1, dis, b1);

    // layer 2
    k_gemm2<<<(ROW_TILES + 3) / 4, 128, 0, stream>>>(agg1, W2, h2);
    k_scatter2<<<(NE * 16) / B, B, 0, stream>>>(src, dst, norm, h2, out);
    k_post2<<<(NN + B - 1) / B, B, 0, stream>>>(out, h2, dis, b2);
}